// XTLoss_9715216023743
// MI455X (gfx1250) — compile-verified
//
#include <hip/hip_runtime.h>

// ---------------------------------------------------------------------------
// XTLoss (stereo photometric LCN loss) for MI455X / gfx1250.
// Memory-bound pipeline: warp+moments pass, fused (hbox+vbox+loss) tiled pass
// with WMMA-based wave reduction, then a tiny double-precision final reduce.
// ---------------------------------------------------------------------------

namespace {
constexpr int NB   = 8;
constexpr int CH   = 3;
constexpr int HH   = 720;
constexpr int WW   = 1280;
constexpr int KSZ  = 9;
constexpr int PAD  = 4;        // KSZ/2
constexpr int TILE = 16;
constexpr int BLK  = 256;      // 8 waves of 32
constexpr int NHW  = NB * HH * WW;        // 7,372,800
constexpr int NCHW = CH * NHW;            // 22,118,400
}

typedef __attribute__((ext_vector_type(2))) float v2f;
typedef __attribute__((ext_vector_type(8))) float v8f;

// Bilinear warp of right image at (i, j - d), zeros padding, replicating the
// reference's exact f32 formula order so sampling coords match bit-for-bit.
__device__ __forceinline__ void warp3(const float* __restrict__ right,
                                      int n, int i, int j, float d,
                                      float out[3]) {
  float gx = (float)(2 * j + 1) / (float)WW - 1.0f - d * 2.0f / (float)WW;
  float gy = (float)(2 * i + 1) / (float)HH - 1.0f;
  float ix = ((gx + 1.0f) * (float)WW - 1.0f) * 0.5f;
  float iy = ((gy + 1.0f) * (float)HH - 1.0f) * 0.5f;
  float x0f = floorf(ix), y0f = floorf(iy);
  float wx1 = ix - x0f, wy1 = iy - y0f;
  float x1f = x0f + 1.0f, y1f = y0f + 1.0f;
  bool vx0 = (x0f >= 0.0f) && (x0f < (float)WW);
  bool vx1 = (x1f >= 0.0f) && (x1f < (float)WW);
  bool vy0 = (y0f >= 0.0f) && (y0f < (float)HH);
  bool vy1 = (y1f >= 0.0f) && (y1f < (float)HH);
  int x0 = (int)x0f; x0 = x0 < 0 ? 0 : (x0 > WW - 1 ? WW - 1 : x0);
  int x1 = (int)x1f; x1 = x1 < 0 ? 0 : (x1 > WW - 1 ? WW - 1 : x1);
  int y0 = (int)y0f; y0 = y0 < 0 ? 0 : (y0 > HH - 1 ? HH - 1 : y0);
  int y1 = (int)y1f; y1 = y1 < 0 ? 0 : (y1 > HH - 1 ? HH - 1 : y1);
  float w00 = (1.0f - wy1) * (1.0f - wx1);
  float w01 = (1.0f - wy1) * wx1;
  float w10 = wy1 * (1.0f - wx1);
  float w11 = wy1 * wx1;
#pragma unroll
  for (int c = 0; c < CH; ++c) {
    const float* img = right + ((n * CH + c) * HH) * WW;
    float t00 = (vy0 && vx0) ? img[y0 * WW + x0] : 0.0f;
    float t01 = (vy0 && vx1) ? img[y0 * WW + x1] : 0.0f;
    float t10 = (vy1 && vx0) ? img[y1 * WW + x0] : 0.0f;
    float t11 = (vy1 && vx1) ? img[y1 * WW + x1] : 0.0f;
    out[c] = t00 * w00 + t01 * w01 + t10 * w10 + t11 * w11;
  }
}

// Pass 1: warp right image; emit recon (optional) and 4 channel-moment planes
// planes[0]=sum(left_c), [1]=sum(left_c^2), [2]=sum(recon_c), [3]=sum(recon_c^2)
__global__ __launch_bounds__(BLK) void k_prep(
    const float* __restrict__ left, const float* __restrict__ right,
    const float* __restrict__ disp, float* __restrict__ planes,
    float* __restrict__ recon, int store_recon) {
  int idx = blockIdx.x * BLK + threadIdx.x;          // 0..NHW-1 exactly
  int n = idx / (HH * WW);
  int rem = idx - n * (HH * WW);
  int i = rem / WW;
  int j = rem - i * WW;
  float r3[3];
  warp3(right, n, i, j, disp[idx], r3);
  float sL = 0.f, qL = 0.f, sR = 0.f, qR = 0.f;
#pragma unroll
  for (int c = 0; c < CH; ++c) {
    int o = ((n * CH + c) * HH + i) * WW + j;
    float l = left[o];
    if (store_recon) recon[o] = r3[c];
    sL += l;      qL += l * l;
    sR += r3[c];  qR += r3[c] * r3[c];
  }
  planes[idx]            = sL;
  planes[NHW + idx]      = qL;
  planes[2 * NHW + idx]  = sR;
  planes[3 * NHW + idx]  = qR;
}

// Pass 2: per 16x16 tile -- load 24x24 halo of 4 planes to LDS, horizontal
// 9-tap in LDS, vertical 9-tap per thread, LCN + per-pixel loss, then a
// layout-invariant WMMA ones-reduction (sum of all D entries == 16 * sum(A))
// followed by an LDS tree reduce.  All blocks are full, no divergence before
// the WMMA, so EXEC is all ones as the ISA requires.
__global__ __launch_bounds__(BLK) void k_loss(
    const float* __restrict__ left, const float* __restrict__ right,
    const float* __restrict__ disp, const float* __restrict__ recon,
    const float* __restrict__ planes, float* __restrict__ partials,
    int use_recon) {
  constexpr int RT = TILE + 2 * PAD;  // 24
  __shared__ float raw[4 * RT * RT];      // 9216 B
  __shared__ float hb[4 * RT * TILE];     // 6144 B
  __shared__ float red[BLK];              // 1024 B
  const int tpr = WW / TILE;  // 80
  const int tpc = HH / TILE;  // 45
  int bid = blockIdx.x;
  int n = bid / (tpr * tpc);
  int t = bid - n * (tpr * tpc);
  int ty = t / tpr;
  int tx = t - ty * tpr;
  int yb = ty * TILE, xb = tx * TILE;

  // Load 4 planes' haloed tiles (zero padding outside image).
  for (int k = threadIdx.x; k < 4 * RT * RT; k += BLK) {
    int p = k / (RT * RT);
    int rm = k - p * (RT * RT);
    int r = rm / RT;
    int cc = rm - r * RT;
    int y = yb - PAD + r;
    int x = xb - PAD + cc;
    float v = 0.0f;
    if (y >= 0 && y < HH && x >= 0 && x < WW)
      v = planes[p * NHW + (n * HH + y) * WW + x];
    raw[k] = v;
  }
  __syncthreads();

  // Horizontal 9-tap box sums into hb (4 planes x 24 rows x 16 cols).
  for (int k = threadIdx.x; k < 4 * RT * TILE; k += BLK) {
    int p = k / (RT * TILE);
    int rm = k - p * (RT * TILE);
    int r = rm / TILE;
    int cc = rm - r * TILE;
    const float* src = &raw[(p * RT + r) * RT + cc];
    float a = 0.0f;
#pragma unroll
    for (int u = 0; u < KSZ; ++u) a += src[u];
    hb[k] = a;
  }
  __syncthreads();

  int lr = threadIdx.x >> 4;       // 0..15 row in tile
  int lc = threadIdx.x & 15;       // 0..15 col in tile
  float S[4];
#pragma unroll
  for (int p = 0; p < 4; ++p) {
    float a = 0.0f;
#pragma unroll
    for (int u = 0; u < KSZ; ++u) a += hb[(p * RT + lr + u) * TILE + lc];
    S[p] = a;
  }
  float meanL = S[0] / 81.0f;
  float msqL  = S[1] / 81.0f;
  float meanR = S[2] / 81.0f;
  float msqR  = S[3] / 81.0f;
  float stdL = (msqL - meanL * meanL) * 81.0f / 80.0f;
  float stdR = (msqR - meanR * meanR) * 81.0f / 80.0f;
  float dL = stdL + 1e-6f;
  float dR = stdR + 1e-6f;

  int i = yb + lr, j = xb + lc;
  float rv[3];
  if (use_recon) {
#pragma unroll
    for (int c = 0; c < CH; ++c)
      rv[c] = recon[((n * CH + c) * HH + i) * WW + j];
  } else {
    warp3(right, n, i, j, disp[n * HH * WW + i * WW + j], rv);
  }
  float val = 0.0f;
#pragma unroll
  for (int c = 0; c < CH; ++c) {
    float l = left[((n * CH + c) * HH + i) * WW + j];
    float lcnL = (l - meanL) / dL;
    float lcnR = (rv[c] - meanR) / dR;
    val += fabsf((lcnL - lcnR) * stdL);
  }

  // Wave-level WMMA ones-reduction.  B == all-ones is layout independent, so
  // sum over all 256 D entries = 16 * sum over all A entries = 16 * wave sum,
  // for ANY A/D VGPR striping.  Each lane sums its 8 D registers; the LDS
  // tree then sums all lanes of all waves; final scale by 1/16 is exact.
  v2f av; av.x = val;  av.y = 0.0f;
  v2f bv; bv.x = 1.0f; bv.y = 1.0f;
  v8f cv = {};
  cv = __builtin_amdgcn_wmma_f32_16x16x4_f32(
      /*neg_a=*/false, av, /*neg_b=*/false, bv,
      /*c_mod=*/(short)0, cv, /*reuse_a=*/false, /*reuse_b=*/false);
  float s8 = ((cv[0] + cv[1]) + (cv[2] + cv[3])) +
             ((cv[4] + cv[5]) + (cv[6] + cv[7]));
  red[threadIdx.x] = s8;
  __syncthreads();
#pragma unroll
  for (int s = BLK / 2; s > 0; s >>= 1) {
    if (threadIdx.x < s) red[threadIdx.x] += red[threadIdx.x + s];
    __syncthreads();
  }
  if (threadIdx.x == 0) partials[bid] = red[0] * 0.0625f;  // /16 exact
}

// Pass 3: sum 28,800 block partials with a double accumulator -> scalar mean.
__global__ __launch_bounds__(BLK) void k_final(
    const float* __restrict__ partials, int nparts, float* __restrict__ out) {
  __shared__ double dred[BLK];
  double acc = 0.0;
  for (int i = threadIdx.x; i < nparts; i += BLK) acc += (double)partials[i];
  dred[threadIdx.x] = acc;
  __syncthreads();
  for (int s = BLK / 2; s > 0; s >>= 1) {
    if (threadIdx.x < s) dred[threadIdx.x] += dred[threadIdx.x + s];
    __syncthreads();
  }
  if (threadIdx.x == 0) out[0] = (float)(dred[0] / (double)NCHW);
}

extern "C" void kernel_launch(void* const* d_in, const int* in_sizes, int n_in,
                              void* d_out, int out_size, void* d_ws,
                              size_t ws_size, hipStream_t stream) {
  (void)in_sizes; (void)n_in; (void)out_size;
  const float* left  = (const float*)d_in[0];
  const float* right = (const float*)d_in[1];
  const float* disp  = (const float*)d_in[2];

  const int nblkL = NB * (HH / TILE) * (WW / TILE);  // 28800

  float* planes   = (float*)d_ws;                    // 4 * NHW floats
  float* recon    = planes + (size_t)4 * NHW;        // NCHW floats (optional)
  size_t need_full =
      ((size_t)4 * NHW + (size_t)NCHW + (size_t)nblkL) * sizeof(float);
  int use_recon = (ws_size >= need_full) ? 1 : 0;    // deterministic per run
  float* partials = use_recon ? (recon + (size_t)NCHW) : recon;

  k_prep<<<NHW / BLK, BLK, 0, stream>>>(left, right, disp, planes, recon,
                                        use_recon);
  k_loss<<<nblkL, BLK, 0, stream>>>(left, right, disp, recon, planes, partials,
                                    use_recon);
  k_final<<<1, BLK, 0, stream>>>(partials, nblkL, (float*)d_out);
}